// CaptionModel_69501160784341
// MI455X (gfx1250) — compile-verified
//
#include <hip/hip_runtime.h>
#include <math.h>

#define BEAMS 8
#define HDIM  1024
#define VOCAB 30000
#define NCONV 3600
#define NROI  100
#define TLEN  20
#define CTX_CHUNKS 30
#define CTX_ROWS   (NCONV / CTX_CHUNKS)   // 120

typedef float v2f __attribute__((ext_vector_type(2)));
typedef float v8f __attribute__((ext_vector_type(8)));

// ---------------------------------------------------------------------------
// Fast tanh: gfx1250 has V_TANH_F32 (TRANS op).  Per ISA 7.4, a TRANS result
// needs 1 independent instruction before use -> trailing v_nop in asm path.
// ---------------------------------------------------------------------------
__device__ __forceinline__ float fast_tanhf(float x) {
#if __has_builtin(__builtin_amdgcn_tanhf)
  return __builtin_amdgcn_tanhf(x);
#elif __has_builtin(__builtin_amdgcn_tanh_f32)
  return __builtin_amdgcn_tanh_f32(x);
#else
  float r;
  asm volatile("v_tanh_f32 %0, %1\n\tv_nop" : "=v"(r) : "v"(x));
  return r;
#endif
}

// ---------------------------------------------------------------------------
// WMMA f32 16x16x4 dual-tile K-loop.  A is [16 x lda] (rows 8..15 zero-pad),
// B row-major [K x ldb].  Ap points at A[m][kh], Bp at B[kh][n0+m].
// Two independent accumulators interleave to hide WMMA D->C latency; pointer
// induction avoids per-load 64-bit address rebuilds.
// ---------------------------------------------------------------------------
__device__ __forceinline__ void wmma2_block(v8f& acc0, v8f& acc1,
                                            const float* __restrict__ Ap,
                                            const float* __restrict__ Bp,
                                            int ldb, int K) {
#pragma unroll 2
  for (int k = 0; k < K; k += 4) {
    v2f a, b0, b1;
    a.x  = Ap[0];  a.y  = Ap[1];
    b0.x = Bp[0];  b0.y = Bp[ldb];
    b1.x = Bp[16]; b1.y = Bp[ldb + 16];
    acc0 = __builtin_amdgcn_wmma_f32_16x16x4_f32(false, a, false, b0, (short)0, acc0, false, false);
    acc1 = __builtin_amdgcn_wmma_f32_16x16x4_f32(false, a, false, b1, (short)0, acc1, false, false);
    Ap += 4; Bp += (size_t)4 * ldb;
  }
}

__device__ __forceinline__ void wmma1_block(v8f& acc0,
                                            const float* __restrict__ Ap,
                                            const float* __restrict__ Bp,
                                            int ldb, int K) {
#pragma unroll 2
  for (int k = 0; k < K; k += 4) {
    v2f a, b0;
    a.x  = Ap[0]; a.y  = Ap[1];
    b0.x = Bp[0]; b0.y = Bp[ldb];
    acc0 = __builtin_amdgcn_wmma_f32_16x16x4_f32(false, a, false, b0, (short)0, acc0, false, false);
    Ap += 4; Bp += (size_t)4 * ldb;
  }
}

// D[8 x N] = A[16(pad) x K] * B[K x N] (+ bias).  One wave per 32 columns
// (2 tiles); launched as 1-wave blocks for wide WGP spread (bandwidth-bound).
__global__ void wmma_gemm8(const float* __restrict__ A, const float* __restrict__ B,
                           const float* __restrict__ bias, float* __restrict__ D,
                           int K, int N) {
  int wave = (int)((blockIdx.x * blockDim.x + threadIdx.x) >> 5);
  int lane = (int)(threadIdx.x & 31u);
  int n0 = wave * 32;
  if (n0 >= N) return;                    // wave-uniform
  int m  = lane & 15;
  int kh = (lane >> 4) << 1;
  const float* Ap = A + (size_t)m * K + kh;
  const float* Bp = B + (size_t)kh * N + n0 + m;
  v8f acc0 = {}, acc1 = {};
  bool two = (n0 + 32 <= N);              // N assumed multiple of 16
  if (two) wmma2_block(acc0, acc1, Ap, Bp, N, K);
  else     wmma1_block(acc0, Ap, Bp, N, K);
  if (lane < 16) {
    float bv0 = bias ? bias[n0 + lane] : 0.f;
#pragma unroll
    for (int r = 0; r < 8; ++r) D[(size_t)r * N + n0 + lane] = acc0[r] + bv0;
    if (two) {
      float bv1 = bias ? bias[n0 + 16 + lane] : 0.f;
#pragma unroll
      for (int r = 0; r < 8; ++r) D[(size_t)r * N + n0 + 16 + lane] = acc1[r] + bv1;
    }
  }
}

// z = [xt | ctx] @ W_lstm + h @ U_lstm + b  -> [8 x 4096]
__global__ void wmma_lstm_z(const float* __restrict__ xt, const float* __restrict__ ctx,
                            const float* __restrict__ hg, const float* __restrict__ W,
                            const float* __restrict__ U, const float* __restrict__ bias,
                            float* __restrict__ Z) {
  const int N = 4 * HDIM;
  int wave = (int)((blockIdx.x * blockDim.x + threadIdx.x) >> 5);
  int lane = (int)(threadIdx.x & 31u);
  int n0 = wave * 32;
  if (n0 >= N) return;
  int m  = lane & 15;
  int kh = (lane >> 4) << 1;
  v8f acc0 = {}, acc1 = {};
  wmma2_block(acc0, acc1, xt  + (size_t)m * HDIM + kh, W + (size_t)kh * N + n0 + m,          N, HDIM);
  wmma2_block(acc0, acc1, ctx + (size_t)m * HDIM + kh, W + (size_t)(HDIM + kh) * N + n0 + m, N, HDIM);
  wmma2_block(acc0, acc1, hg  + (size_t)m * HDIM + kh, U + (size_t)kh * N + n0 + m,          N, HDIM);
  if (lane < 16) {
    float bv0 = bias[n0 + lane], bv1 = bias[n0 + 16 + lane];
#pragma unroll
    for (int r = 0; r < 8; ++r) {
      Z[(size_t)r * N + n0 + lane]      = acc0[r] + bv0;
      Z[(size_t)r * N + n0 + 16 + lane] = acc1[r] + bv1;
    }
  }
}

// ---------------------------------------------------------------------------
// Two-scan log_softmax + top-8.  Scan 1: per-thread register top-8 (branchless
// compare-swap insertion, composite key (value desc, index asc)), merged via a
// fixed-index bitonic merge network through LDS.  Global max = top-1, so no
// separate max pass.  Scan 2: sum(exp(x-max)).
// ---------------------------------------------------------------------------
__device__ __forceinline__ void merge8(float* tv, int* ti,
                                       const float* ov, const int* oi) {
  float x[16]; int xi[16];
#pragma unroll
  for (int u = 0; u < 8; ++u) {
    x[u] = tv[u];         xi[u] = ti[u];
    x[8 + u] = ov[7 - u]; xi[8 + u] = oi[7 - u];   // reverse -> bitonic
  }
#define CSW(i, j)                                                         \
  { bool sw = (x[j] > x[i]) || (x[j] == x[i] && xi[j] < xi[i]);           \
    float fv = sw ? x[j] : x[i]; int fi = sw ? xi[j] : xi[i];             \
    x[j] = sw ? x[i] : x[j];     xi[j] = sw ? xi[i] : xi[j];              \
    x[i] = fv;                   xi[i] = fi; }
  CSW(0, 8) CSW(1, 9) CSW(2, 10) CSW(3, 11) CSW(4, 12) CSW(5, 13) CSW(6, 14) CSW(7, 15)
  CSW(0, 4) CSW(1, 5) CSW(2, 6)  CSW(3, 7)
  CSW(0, 2) CSW(1, 3) CSW(4, 6)  CSW(5, 7)
  CSW(0, 1) CSW(2, 3) CSW(4, 5)  CSW(6, 7)
#undef CSW
#pragma unroll
  for (int u = 0; u < 8; ++u) { tv[u] = x[u]; ti[u] = xi[u]; }
}

__global__ void logsoftmax_top8(const float* __restrict__ logits,
                                float* __restrict__ topv, int* __restrict__ topi) {
  int b = blockIdx.x;
  const float* row = logits + (size_t)b * VOCAB;
  int tid = threadIdx.x;
  __shared__ float sv[256][8];
  __shared__ int   si[256][8];
  __shared__ float red[256];
  __shared__ float s_gv[8];
  __shared__ int   s_gi[8];
  __shared__ float s_lse;

  float tv[8]; int ti[8];
#pragma unroll
  for (int u = 0; u < 8; ++u) { tv[u] = -INFINITY; ti[u] = 0x7fffffff; }
  for (int j = tid; j < VOCAB; j += 256) {
    float v = row[j];
    if (v > tv[7]) {
      float cv = v; int ci = j;
#pragma unroll
      for (int u = 0; u < 8; ++u) {
        bool sw = (cv > tv[u]) || (cv == tv[u] && ci < ti[u]);
        float nv = sw ? cv : tv[u]; int ni = sw ? ci : ti[u];
        cv = sw ? tv[u] : cv;       ci = sw ? ti[u] : ci;
        tv[u] = nv; ti[u] = ni;
      }
    }
  }
#pragma unroll
  for (int u = 0; u < 8; ++u) { sv[tid][u] = tv[u]; si[tid][u] = ti[u]; }
  __syncthreads();
  for (int s = 128; s > 0; s >>= 1) {
    if (tid < s) {
      float ov[8]; int oi[8];
#pragma unroll
      for (int u = 0; u < 8; ++u) { ov[u] = sv[tid + s][u]; oi[u] = si[tid + s][u]; }
      merge8(tv, ti, ov, oi);
#pragma unroll
      for (int u = 0; u < 8; ++u) { sv[tid][u] = tv[u]; si[tid][u] = ti[u]; }
    }
    __syncthreads();
  }
  if (tid == 0) {
#pragma unroll
    for (int u = 0; u < 8; ++u) { s_gv[u] = tv[u]; s_gi[u] = ti[u]; }
  }
  __syncthreads();
  float mx = s_gv[0];
  float sum = 0.f;
  for (int j = tid; j < VOCAB; j += 256) sum += expf(row[j] - mx);
  red[tid] = sum; __syncthreads();
  for (int s = 128; s > 0; s >>= 1) { if (tid < s) red[tid] += red[tid + s]; __syncthreads(); }
  if (tid == 0) s_lse = logf(red[0]);
  __syncthreads();
  if (tid == 0) {
#pragma unroll
    for (int u = 0; u < 8; ++u) {
      topi[b * 8 + u] = s_gi[u];
      topv[b * 8 + u] = s_gv[u] - mx - s_lse;
    }
  }
}

// ---------------------------------------------------------------------------
// Single-block beam merge: candidate top-8 over 64, beam-state reorder,
// snapshots, pointer-mask scatter, embedding gather.
// ---------------------------------------------------------------------------
__global__ void beam_merge(int t,
                           const float* __restrict__ topv, const int* __restrict__ topi,
                           float* __restrict__ bls,
                           const int* __restrict__ seq_in, int* __restrict__ seq_out,
                           const float* __restrict__ lps_in, float* __restrict__ lps_out,
                           int* __restrict__ seq_snap, float* __restrict__ lps_snap,
                           float* __restrict__ ds_row,
                           const float* __restrict__ h_in, float* __restrict__ h_out,
                           const float* __restrict__ c_in, float* __restrict__ c_out,
                           const float* __restrict__ pm_in, float* __restrict__ pm_out,
                           const int* __restrict__ roi_offset,
                           const float* __restrict__ embed_W, float* __restrict__ xt) {
  __shared__ int   sq[8], stok[8];
  __shared__ float slocal[8];
  int tid = threadIdx.x;
  if (tid == 0) {
    float cand[64];
    for (int r = 0; r < 8; ++r)
      for (int c = 0; c < 8; ++c) {
        float v = bls[r] + topv[r * 8 + c];
        if (t == 0 && r > 0) v = -INFINITY;
        cand[r * 8 + c] = v;
      }
    bool used[64];
    for (int f = 0; f < 64; ++f) used[f] = false;
    for (int s = 0; s < 8; ++s) {
      int bi = -1; float bv = 0.f;
      for (int f = 0; f < 64; ++f)
        if (!used[f] && (bi < 0 || cand[f] > bv)) { bi = f; bv = cand[f]; }
      used[bi] = true;
      int q = bi >> 3, col = bi & 7;
      sq[s]     = q;
      stok[s]   = topi[q * 8 + col];
      slocal[s] = topv[q * 8 + col];
      bool done = (stok[s] == 0) || (t == TLEN - 1);
      ds_row[s] = done ? bv : -INFINITY;
      bls[s]    = done ? -1000.f : bv;
    }
  }
  __syncthreads();
  for (int i = tid; i < TLEN * 8; i += blockDim.x) {
    int r = i / 8, s = i % 8;
    int   svv = (r == t) ? stok[s]   : seq_in[r * 8 + sq[s]];
    float lv  = (r == t) ? slocal[s] : lps_in[r * 8 + sq[s]];
    seq_out[i] = svv; lps_out[i] = lv;
    seq_snap[i] = svv; lps_snap[i] = lv;
  }
  for (int i = tid; i < 8 * HDIM; i += blockDim.x) {
    int s = i / HDIM, j = i % HDIM;
    h_out[i] = h_in[sq[s] * HDIM + j];
    c_out[i] = c_in[sq[s] * HDIM + j];
  }
  for (int i = tid; i < 8 * (NROI + 1); i += blockDim.x) {
    int s = i / (NROI + 1), j = i % (NROI + 1);
    pm_out[i] = pm_in[sq[s] * (NROI + 1) + j];
  }
  __syncthreads();
  if (tid == 0) {
    for (int s = 0; s < 8; ++s) {
      int roi = stok[s] - VOCAB - 1;
      int idx = (roi < 0) ? 0 : (roi + roi_offset[s] + 1);
      if (idx >= 0 && idx < 8 * (NROI + 1)) pm_out[idx] = 1.f;
    }
    pm_out[0] = 0.f;
  }
  __syncthreads();
  for (int i = tid; i < 8 * HDIM; i += blockDim.x) {
    int s = i / HDIM, j = i % HDIM;
    xt[i] = embed_W[(size_t)stok[s] * HDIM + j];
  }
}

// ---------------------------------------------------------------------------
// e[b][n] = dot(tanh(feats[b][n][:] + hW[b][:]), v)  — one wave per (b,n),
// float4 (b128) loads, hardware V_TANH_F32.
// ---------------------------------------------------------------------------
__global__ void att_e(const float* __restrict__ feats, const float* __restrict__ hW,
                      const float* __restrict__ vvec, float* __restrict__ e, int n_loc) {
  int gwave = (int)((blockIdx.x * blockDim.x + threadIdx.x) >> 5);
  int lane  = (int)(threadIdx.x & 31u);
  if (gwave >= 8 * n_loc) return;
  int b = gwave / n_loc, n = gwave % n_loc;
  const float4* row = (const float4*)(feats + ((size_t)b * n_loc + n) * HDIM);
  const float4* hb  = (const float4*)(hW + (size_t)b * HDIM);
  const float4* vv  = (const float4*)vvec;
  float p = 0.f;
#pragma unroll 2
  for (int j = lane; j < HDIM / 4; j += 32) {
    float4 r = row[j], h4 = hb[j], v4 = vv[j];
    p += fast_tanhf(r.x + h4.x) * v4.x + fast_tanhf(r.y + h4.y) * v4.y
       + fast_tanhf(r.z + h4.z) * v4.z + fast_tanhf(r.w + h4.w) * v4.w;
  }
  for (int off = 16; off > 0; off >>= 1) p += __shfl_xor(p, off, 32);
  if (lane == 0) e[b * n_loc + n] = p;
}

__global__ void softmax_rows(const float* __restrict__ x, float* __restrict__ y, int n) {
  int b = blockIdx.x;
  const float* row = x + (size_t)b * n;
  float* out = y + (size_t)b * n;
  __shared__ float red[256];
  __shared__ float s_stat[2];
  int tid = threadIdx.x;
  float m = -INFINITY;
  for (int j = tid; j < n; j += 256) m = fmaxf(m, row[j]);
  red[tid] = m; __syncthreads();
  for (int s = 128; s > 0; s >>= 1) { if (tid < s) red[tid] = fmaxf(red[tid], red[tid + s]); __syncthreads(); }
  if (tid == 0) s_stat[0] = red[0];
  __syncthreads();
  m = s_stat[0];
  float sum = 0.f;
  for (int j = tid; j < n; j += 256) sum += __expf(row[j] - m);
  red[tid] = sum; __syncthreads();
  for (int s = 128; s > 0; s >>= 1) { if (tid < s) red[tid] += red[tid + s]; __syncthreads(); }
  if (tid == 0) s_stat[1] = 1.f / red[0];
  __syncthreads();
  float inv = s_stat[1];
  for (int j = tid; j < n; j += 256) out[j] = __expf(row[j] - m) * inv;
}

// ctx partials: 8 beams x 30 chunks = 240 WGPs pulling the 118MB conv tensor
// (vs 8 before) -> actually able to saturate HBM/L2 bandwidth.
__global__ void att_ctx_part(const float* __restrict__ conv, const float* __restrict__ alpha,
                             float* __restrict__ part) {
  int blk = blockIdx.x;
  int b = blk / CTX_CHUNKS, ch = blk % CTX_CHUNKS;
  int j = threadIdx.x;   // 0..255 float4 lanes
  const float4* base = (const float4*)(conv + ((size_t)b * NCONV + (size_t)ch * CTX_ROWS) * HDIM) + j;
  const float* al = alpha + (size_t)b * NCONV + ch * CTX_ROWS;
  float4 acc = {0.f, 0.f, 0.f, 0.f};
#pragma unroll 4
  for (int n = 0; n < CTX_ROWS; ++n) {
    float a = al[n];
    float4 r = base[(size_t)n * (HDIM / 4)];
    acc.x += a * r.x; acc.y += a * r.y; acc.z += a * r.z; acc.w += a * r.w;
  }
  ((float4*)(part + ((size_t)b * CTX_CHUNKS + ch) * HDIM))[j] = acc;
}

__global__ void att_ctx_reduce(const float* __restrict__ part, float* __restrict__ ctx) {
  int b = blockIdx.x, j = threadIdx.x;
  float4 acc = {0.f, 0.f, 0.f, 0.f};
  for (int ch = 0; ch < CTX_CHUNKS; ++ch) {
    float4 r = ((const float4*)(part + ((size_t)b * CTX_CHUNKS + ch) * HDIM))[j];
    acc.x += r.x; acc.y += r.y; acc.z += r.z; acc.w += r.w;
  }
  ((float4*)(ctx + (size_t)b * HDIM))[j] = acc;
}

__global__ void lstm_gates(const float* __restrict__ z, float* __restrict__ c,
                           float* __restrict__ h, float* __restrict__ ro) {
  int idx = blockIdx.x * blockDim.x + threadIdx.x;
  if (idx >= 8 * HDIM) return;
  int b = idx / HDIM, j = idx % HDIM;
  const float* zr = z + (size_t)b * 4 * HDIM;
  float ig = 1.f / (1.f + expf(-zr[j]));
  float fg = 1.f / (1.f + expf(-zr[HDIM + j]));
  float og = 1.f / (1.f + expf(-zr[2 * HDIM + j]));
  float gg = tanhf(zr[3 * HDIM + j]);
  float cn = fg * c[idx] + ig * gg;
  float hn = og * tanhf(cn);
  c[idx] = cn; h[idx] = hn; ro[idx] = hn;
}

__global__ void att2_argmax(const float* __restrict__ e2, const float* __restrict__ pm,
                            int* __restrict__ att2_ind) {
  int b = threadIdx.x;
  if (b < 8) {
    float best = -INFINITY; int bi = 0;
    for (int r = 0; r < NROI; ++r) {
      float v = e2[b * NROI + r];
      if (pm[b * (NROI + 1) + 1 + r] > 0.5f) v = -1e9f;
      if (v > best) { best = v; bi = r; }
    }
    att2_ind[b] = bi;
  }
}

__global__ void init_state(const float* __restrict__ ro0, const float* __restrict__ h0,
                           const float* __restrict__ c0, const float* __restrict__ pm0,
                           float* roP0, float* roP1, float* hP0, float* hP1, float* cP0,
                           float* pmP0, float* bls, int* seqP0, float* lpsP0,
                           float* xt, float* ctx) {
  int i = blockIdx.x * blockDim.x + threadIdx.x;
  if (i < 8 * HDIM) { roP0[i] = ro0[i]; hP0[i] = h0[i]; cP0[i] = c0[i]; }
  if (i >= 8 * HDIM && i < 16 * HDIM) {   // zero A-matrix pad rows 8..15
    roP0[i] = 0.f; roP1[i] = 0.f; hP0[i] = 0.f; hP1[i] = 0.f; xt[i] = 0.f; ctx[i] = 0.f;
  }
  if (i < 8 * (NROI + 1)) pmP0[i] = pm0[i];
  if (i < 8) bls[i] = 0.f;
  if (i < TLEN * 8) { seqP0[i] = 0; lpsP0[i] = 0.f; }
}

__global__ void finalize(const float* __restrict__ done_scores,
                         const int* __restrict__ seq_snaps,
                         const float* __restrict__ lps_snaps,
                         float* __restrict__ out) {
  __shared__ int s_s[8], s_b[8];
  if (threadIdx.x == 0) {
    bool used[TLEN * 8];
    for (int f = 0; f < TLEN * 8; ++f) used[f] = false;
    for (int i = 0; i < 8; ++i) {
      int bi = -1; float bv = 0.f;
      for (int f = 0; f < TLEN * 8; ++f)
        if (!used[f] && (bi < 0 || done_scores[f] > bv)) { bi = f; bv = done_scores[f]; }
      used[bi] = true;
      s_s[i] = bi / 8; s_b[i] = bi % 8;
      out[2 * TLEN * 8 + i] = bv;   // best_vals
    }
  }
  __syncthreads();
  int tid = threadIdx.x;
  if (tid < 8 * TLEN) {
    int i = tid / TLEN, r = tid % TLEN;
    out[i * TLEN + r]            = (float)seq_snaps[s_s[i] * (TLEN * 8) + r * 8 + s_b[i]];
    out[TLEN * 8 + i * TLEN + r] = lps_snaps[s_s[i] * (TLEN * 8) + r * 8 + s_b[i]];
  }
}

// ---------------------------------------------------------------------------
extern "C" void kernel_launch(void* const* d_in, const int* in_sizes, int n_in,
                              void* d_out, int out_size, void* d_ws, size_t ws_size,
                              hipStream_t stream) {
  const float* embed_W    = (const float*)d_in[0];
  const float* logit_W    = (const float*)d_in[1];
  const float* logit_b    = (const float*)d_in[2];
  const float* W_lstm     = (const float*)d_in[3];
  const float* U_lstm     = (const float*)d_in[4];
  const float* b_lstm     = (const float*)d_in[5];
  const float* w_att      = (const float*)d_in[6];
  const float* v_att      = (const float*)d_in[7];
  const float* w_att2     = (const float*)d_in[8];
  const float* v_att2     = (const float*)d_in[9];
  const float* conv_feats = (const float*)d_in[10];
  const float* pool_feats = (const float*)d_in[11];
  const float* ro0        = (const float*)d_in[12];
  const float* h0         = (const float*)d_in[13];
  const float* c0         = (const float*)d_in[14];
  const float* pm0        = (const float*)d_in[15];
  const int*   roi_off    = (const int*)d_in[16];
  (void)in_sizes; (void)n_in; (void)out_size; (void)ws_size;

  char* base = (char*)d_ws;
  size_t off = 0;
  auto alloc = [&](size_t bytes) -> void* {
    void* p = base + off;
    off = (off + bytes + 255) & ~(size_t)255;
    return p;
  };

  float* logits    = (float*)alloc(8 * VOCAB * 4);
  float* topv      = (float*)alloc(64 * 4);
  int*   topi      = (int*)  alloc(64 * 4);
  float* bls       = (float*)alloc(8 * 4);
  int*   seqP[2]   = {(int*)alloc(TLEN * 8 * 4), (int*)alloc(TLEN * 8 * 4)};
  float* lpsP[2]   = {(float*)alloc(TLEN * 8 * 4), (float*)alloc(TLEN * 8 * 4)};
  int*   seq_snaps = (int*)  alloc(TLEN * TLEN * 8 * 4);
  float* lps_snaps = (float*)alloc(TLEN * TLEN * 8 * 4);
  float* done_sc   = (float*)alloc(TLEN * 8 * 4);
  float* hP[2]     = {(float*)alloc(16 * HDIM * 4), (float*)alloc(16 * HDIM * 4)};
  float* cP[2]     = {(float*)alloc(8 * HDIM * 4),  (float*)alloc(8 * HDIM * 4)};
  float* roP[2]    = {(float*)alloc(16 * HDIM * 4), (float*)alloc(16 * HDIM * 4)};
  float* pmP[2]    = {(float*)alloc(8 * (NROI + 1) * 4), (float*)alloc(8 * (NROI + 1) * 4)};
  float* xt        = (float*)alloc(16 * HDIM * 4);
  float* ctxb      = (float*)alloc(16 * HDIM * 4);
  float* hW        = (float*)alloc(8 * HDIM * 4);
  float* hw2       = (float*)alloc(8 * HDIM * 4);
  float* e_buf     = (float*)alloc(8 * NCONV * 4);
  float* alpha     = (float*)alloc(8 * NCONV * 4);
  float* ctx_part  = (float*)alloc((size_t)8 * CTX_CHUNKS * HDIM * 4);
  float* z_buf     = (float*)alloc(8 * 4 * HDIM * 4);
  float* e2_buf    = (float*)alloc(8 * NROI * 4);
  int*   att2_ind  = (int*)  alloc(8 * 4);

  init_state<<<64, 256, 0, stream>>>(ro0, h0, c0, pm0, roP[0], roP[1], hP[0], hP[1],
                                     cP[0], pmP[0], bls, seqP[0], lpsP[0], xt, ctxb);

  const int logitWaves = (VOCAB / 16 + 1) / 2;   // 938 (last wave single-tile)
  for (int t = 0; t < TLEN; ++t) {
    int p = t & 1, np = 1 - p;
    // logits = rnn_out @ logit_W + b  (1-wave blocks for max WGP spread)
    wmma_gemm8<<<logitWaves, 32, 0, stream>>>(roP[p], logit_W, logit_b, logits, HDIM, VOCAB);
    logsoftmax_top8<<<8, 256, 0, stream>>>(logits, topv, topi);
    beam_merge<<<1, 256, 0, stream>>>(t, topv, topi, bls,
                                      seqP[p], seqP[np], lpsP[p], lpsP[np],
                                      seq_snaps + t * TLEN * 8, lps_snaps + t * TLEN * 8,
                                      done_sc + t * 8,
                                      hP[p], hP[np], cP[p], cP[np],
                                      pmP[p], pmP[np], roi_off, embed_W, xt);
    // visual attention
    wmma_gemm8<<<32, 32, 0, stream>>>(hP[np], w_att, nullptr, hW, HDIM, HDIM);
    att_e<<<(8 * NCONV) / 8, 256, 0, stream>>>(conv_feats, hW, v_att, e_buf, NCONV);
    softmax_rows<<<8, 256, 0, stream>>>(e_buf, alpha, NCONV);
    att_ctx_part<<<8 * CTX_CHUNKS, 256, 0, stream>>>(conv_feats, alpha, ctx_part);
    att_ctx_reduce<<<8, 256, 0, stream>>>(ctx_part, ctxb);
    // LSTM cell
    wmma_lstm_z<<<128, 32, 0, stream>>>(xt, ctxb, hP[np], W_lstm, U_lstm, b_lstm, z_buf);
    lstm_gates<<<32, 256, 0, stream>>>(z_buf, cP[np], hP[np], roP[np]);
    // pointer attention (faithfulness; result unused by outputs)
    wmma_gemm8<<<32, 32, 0, stream>>>(hP[np], w_att2, nullptr, hw2, HDIM, HDIM);
    att_e<<<(8 * NROI) / 8, 256, 0, stream>>>(pool_feats, hw2, v_att2, e2_buf, NROI);
    att2_argmax<<<1, 64, 0, stream>>>(e2_buf, pmP[np], att2_ind);
  }

  finalize<<<1, 256, 0, stream>>>(done_sc, seq_snaps, lps_snaps, (float*)d_out);
}